// ResAttentionLayer_29257317220720
// MI455X (gfx1250) — compile-verified
//
#include <hip/hip_runtime.h>
#include <math.h>

#define HIDDEN 128
#define ALPHA  0.2f
#define THR    0.5f

// j-chunk of Wh staged into LDS per block iteration, stored K-pair-interleaved:
// element (pair p, col c, u) at whl[p*PS + 2c + u]  (p = j/2 within chunk, u = j&1)
// PS = 288 dwords => pair stride ≡ 32 (mod 64 banks): the two half-waves of a
// wave (reading pairs p and p+1) hit disjoint bank halves -> conflict-free b64 loads.
#define TJ 64
#define PS (2 * HIDDEN + 32)   // 288 dwords per pair-row

typedef __attribute__((ext_vector_type(2))) float v2f;
typedef __attribute__((ext_vector_type(8))) float v8f;

__device__ __forceinline__ float lrelu(float x) { return x > 0.f ? x : ALPHA * x; }

// ---------------------------------------------------------------------------
// Kernel 1: Wh = X @ W^T + b ; s1 = Wh @ a1 ; s2 = Wh @ a2
// One block (128 threads) per row i. W stays hot in L2 (64 KB).
// ---------------------------------------------------------------------------
__global__ __launch_bounds__(HIDDEN) void wh_kernel(
    const float* __restrict__ X, const float* __restrict__ W,
    const float* __restrict__ b, const float* __restrict__ a,
    float* __restrict__ Wh, float* __restrict__ s1, float* __restrict__ s2) {
  __shared__ float xrow[HIDDEN];
  __shared__ float r1[HIDDEN];
  __shared__ float r2[HIDDEN];
  const int i = blockIdx.x;
  const int t = threadIdx.x;
  xrow[t] = X[(size_t)i * HIDDEN + t];
  __syncthreads();
  float acc = b[t];
  const float* __restrict__ wrow = W + (size_t)t * HIDDEN;
#pragma unroll 8
  for (int k = 0; k < HIDDEN; ++k) acc = fmaf(xrow[k], wrow[k], acc);
  Wh[(size_t)i * HIDDEN + t] = acc;
  r1[t] = acc * a[t];
  r2[t] = acc * a[HIDDEN + t];
  __syncthreads();
  for (int off = HIDDEN / 2; off > 0; off >>= 1) {
    if (t < off) { r1[t] += r1[t + off]; r2[t] += r2[t + off]; }
    __syncthreads();
  }
  if (t == 0) { s1[i] = r1[0]; s2[i] = r2[0]; }
}

// ---------------------------------------------------------------------------
// Kernel 2: gmax = max_j s2[j]  (leaky_relu is monotonic -> row max of the
// logit matrix is lrelu(s1_i + gmax); one scalar suffices for all rows)
// ---------------------------------------------------------------------------
__global__ __launch_bounds__(256) void gmax_kernel(
    const float* __restrict__ s2, float* __restrict__ gmax, int N) {
  __shared__ float red[256];
  float m = -1e30f;
  for (int j = threadIdx.x; j < N; j += 256) m = fmaxf(m, s2[j]);
  red[threadIdx.x] = m;
  __syncthreads();
  for (int off = 128; off > 0; off >>= 1) {
    if (threadIdx.x < off)
      red[threadIdx.x] = fmaxf(red[threadIdx.x], red[threadIdx.x + off]);
    __syncthreads();
  }
  if (threadIdx.x == 0) gmax[0] = red[0];
}

// ---------------------------------------------------------------------------
// Kernel 3: invden[i] = 1 / sum_j exp(lrelu(s1_i + s2_j) - m_i)
// ---------------------------------------------------------------------------
__global__ __launch_bounds__(256) void denom_kernel(
    const float* __restrict__ s1, const float* __restrict__ s2,
    const float* __restrict__ gmax, float* __restrict__ invden, int N) {
  __shared__ float red[256];
  const int i = blockIdx.x;
  const float s1v = s1[i];
  const float m = lrelu(s1v + gmax[0]);
  float sum = 0.f;
  for (int j = threadIdx.x; j < N; j += 256)
    sum += __expf(lrelu(s1v + s2[j]) - m);
  red[threadIdx.x] = sum;
  __syncthreads();
  for (int off = 128; off > 0; off >>= 1) {
    if (threadIdx.x < off) red[threadIdx.x] += red[threadIdx.x + off];
    __syncthreads();
  }
  if (threadIdx.x == 0) invden[i] = 1.f / red[0];
}

// ---------------------------------------------------------------------------
// Kernel 4: h' = att @ Wh fused with attention generation + elu epilogue,
// using V_WMMA_F32_16X16X4_F32 (exact fp32 matrix path on CDNA5).
//
// Block = 256 threads = 8 waves; each wave owns a 16-row x 128-col C strip
// (8 accumulator tiles of 16x16 f32, 64 VGPRs). The block cooperatively
// stages TJ=64 rows of Wh (+ s2 chunk) into LDS in K-pair-interleaved form,
// then each wave streams K in steps of 4:
//   1) compute the thresholded-attention A fragment (2 v_exp, overlaps DS)
//   2) bulk-load all 8 B fragments (4x ds_load_2addr_b64, latencies overlap)
//   3) fire 8 back-to-back v_wmma_f32_16x16x4_f32
//   A layout (32b 16x4): lanes 0-15 hold K=0(V0),K=1(V1); lanes 16-31 K=2,K=3
//   B layout (32b 4x16): lanes 0-15 hold rows K=0(V0),K=1(V1); 16-31 K=2,K=3
// ---------------------------------------------------------------------------
__global__ __launch_bounds__(256) void hprime_kernel(
    const float* __restrict__ X, const float* __restrict__ Wh,
    const float* __restrict__ s1, const float* __restrict__ s2,
    const float* __restrict__ invden, const float* __restrict__ gmax,
    float* __restrict__ out, int N) {
  __shared__ float whl[(TJ / 2) * PS];   // ~36 KB
  __shared__ float s2l[TJ];

  const int tid  = threadIdx.x;
  const int lane = tid & 31;
  const int wave = tid >> 5;
  const int i0   = blockIdx.x * 128 + wave * 16;  // this wave's 16-row tile
  const int rA   = lane & 15;                     // row within tile / col-in-16
  const int jsel = lane >> 4;                     // which K pair this half-wave holds

  const int   row  = i0 + rA;
  const float s1v  = s1[row];
  const float m    = lrelu(s1v + gmax[0]);
  const float invd = invden[row];

  v8f acc[8] = {};

  for (int jb = 0; jb < N; jb += TJ) {
    // ---- stage Wh[jb .. jb+TJ) pair-interleaved + s2 chunk into LDS ----
    // thread handles (pair p, col c): coalesced global reads (consecutive c),
    // stride-2-dword ds_store_b64 (32 lanes cover all 64 banks once).
    for (int idx = tid; idx < (TJ / 2) * HIDDEN; idx += 256) {
      const int c = idx & (HIDDEN - 1);
      const int p = idx >> 7;                       // HIDDEN == 128
      const size_t gbase = (size_t)(jb + 2 * p) * HIDDEN + c;
      float2 pr;
      pr.x = Wh[gbase];
      pr.y = Wh[gbase + HIDDEN];
      *(float2*)(whl + p * PS + 2 * c) = pr;
    }
    if (tid < TJ) s2l[tid] = s2[jb + tid];
    __syncthreads();

    // running LDS base pointer for this half-wave's K pair (strength-reduced)
    const float* bbase = whl + jsel * PS + 2 * rA;
    for (int jr = 0; jr < TJ; jr += 4) {
      // (1) A fragment: thresholded softmax weights for this 16x4 tile
      v2f afrag;
#pragma unroll
      for (int u = 0; u < 2; ++u) {
        const int k = jsel * 2 + u;
        const float e = lrelu(s1v + s2l[jr + k]);
        float att = __expf(e - m) * invd;
        afrag[u] = (att < THR) ? att : 0.f;
      }
      // (2) bulk-load all 8 B fragments (one aligned b64 each; compiler pairs
      //     them into ds_load_2addr_b64) so their latencies overlap
      v2f bfr[8];
#pragma unroll
      for (int nt = 0; nt < 8; ++nt)
        bfr[nt] = *(const v2f*)(bbase + 32 * nt);
      // (3) 8 back-to-back WMMAs sharing the A fragment
#pragma unroll
      for (int nt = 0; nt < 8; ++nt)
        acc[nt] = __builtin_amdgcn_wmma_f32_16x16x4_f32(
            false, afrag, false, bfr[nt], (short)0, acc[nt],
            /*reuse_a=*/false, /*reuse_b=*/false);
      bbase += 2 * PS;
    }
    __syncthreads();
  }

  // ---- epilogue: out = elu(X - h'), straight from C-fragment layout ----
  // C layout: VGPR v -> row (v) for lanes 0-15, row (v+8) for lanes 16-31;
  //           lane%16 -> column within the 16-wide tile.
#pragma unroll
  for (int nt = 0; nt < 8; ++nt) {
#pragma unroll
    for (int v = 0; v < 8; ++v) {
      const int r = i0 + v + jsel * 8;
      const int c = nt * 16 + rA;
      const float x = X[(size_t)r * HIDDEN + c];
      const float d = x - acc[nt][v];
      out[(size_t)r * HIDDEN + c] = d > 0.f ? d : expm1f(d);
    }
  }
}

// ---------------------------------------------------------------------------
extern "C" void kernel_launch(void* const* d_in, const int* in_sizes, int n_in,
                              void* d_out, int out_size, void* d_ws, size_t ws_size,
                              hipStream_t stream) {
  const float* X = (const float*)d_in[0];  // entity_embs (N, 128)
  const float* W = (const float*)d_in[1];  // (128, 128)
  const float* b = (const float*)d_in[2];  // (128,)
  const float* a = (const float*)d_in[3];  // (256, 1) flat
  float* out = (float*)d_out;

  const int N = in_sizes[0] / HIDDEN;      // 8192

  // workspace layout (f32): Wh | s1 | s2 | invden | gmax  (~4.2 MB)
  float* ws     = (float*)d_ws;
  float* Wh     = ws;
  float* s1     = Wh + (size_t)N * HIDDEN;
  float* s2     = s1 + N;
  float* invden = s2 + N;
  float* gmax   = invden + N;

  wh_kernel    <<<N,       HIDDEN, 0, stream>>>(X, W, b, a, Wh, s1, s2);
  gmax_kernel  <<<1,       256,    0, stream>>>(s2, gmax, N);
  denom_kernel <<<N,       256,    0, stream>>>(s1, s2, gmax, invden, N);
  hprime_kernel<<<N / 128, 256,    0, stream>>>(X, Wh, s1, s2, invden, gmax, out, N);
}